// Ricker_Predation_3839700762670
// MI455X (gfx1250) — compile-verified
//
#include <hip/hip_runtime.h>
#include <hip/hip_bf16.h>

// Ricker predation 2-state nonlinear scan, T_LEN = 1<<20.
// Inherently serial recurrence -> single-wave latency-optimized kernel:
//   - 32 lanes cooperatively precompute state-independent coefficients into LDS
//   - 32 lanes prefetch the next tile's global lines (global_prefetch_b8)
//   - lane 0 runs the serial chain: fma -> fma -> v_exp_f32 -> fma per state
// log2(e) folded into all coefficients so exp() is a single native v_exp_f32.

#define RP_TILE 256  // steps per tile (LDS: 256 * 16B = 4KB)

__global__ __launch_bounds__(32) void ricker_scan_kernel(
    const float* __restrict__ Temp,
    const float* __restrict__ params,
    const float* __restrict__ eps,
    float* __restrict__ out,
    int n)  // n == T_LEN
{
    const int lane = threadIdx.x;
    const int NT = n - 1;  // number of scan steps

    __shared__ float4 coef[RP_TILE];  // (p1, p2, ns*e1, ns*e2) per step

    // ---- uniform parameter load + coefficient folding (off the hot loop) ----
    const float L2E = 1.44269504088896340736f;  // log2(e)
    const float a1 = params[0], b1 = params[1], g1 = params[2],
                bx1 = params[3], cx1 = params[4];
    const float a2 = params[5], b2 = params[6], g2 = params[7],
                bx2 = params[8], cx2 = params[9];
    const float sigma = params[10];

    // exp(a*(1 - b*x - g*y + bx*t + cx*t^2)) == exp2(k + kb*t + kc*t^2 + ab*x + ag*y)
    const float k1 = L2E * a1, kb1 = L2E * a1 * bx1, kc1 = L2E * a1 * cx1;
    const float ab1 = -L2E * a1 * b1, ag1 = -L2E * a1 * g1;
    const float k2 = L2E * a2, kb2 = L2E * a2 * bx2, kc2 = L2E * a2 * cx2;
    const float ab2 = -L2E * a2 * b2, ag2 = -L2E * a2 * g2;
    const float ns = sigma * 0.1f;

    const float* __restrict__ e1p = eps;       // noise_eps[0], NT elements
    const float* __restrict__ e2p = eps + NT;  // noise_eps[1], NT elements

    float x = 1.0f, y = 1.0f;  // carried state lives in lane 0's registers
    if (lane == 0) {
        out[0] = 1.0f;   // x_traj[0]
        out[n] = 1.0f;   // y_traj[0]
    }

    for (int base = 0; base < NT; base += RP_TILE) {
        const int nsteps = (NT - base < RP_TILE) ? (NT - base) : RP_TILE;

        // ---- phase 1: wave-cooperative precompute of state-independent terms ----
        #pragma unroll
        for (int j = lane; j < RP_TILE; j += 32) {
            const int i = base + j;
            if (j < nsteps) {
                const float t = Temp[i];
                const float p1 = fmaf(kc1, t * t, fmaf(kb1, t, k1));
                const float p2 = fmaf(kc2, t * t, fmaf(kb2, t, k2));
                coef[j] = make_float4(p1, p2, ns * e1p[i], ns * e2p[i]);
            }
        }

        // ---- phase 1b: prefetch next tile's lines into near cache (gfx1250
        // global_prefetch_b8). 16 lines x 64B = 1KB = one tile per stream. ----
        const int nb = base + RP_TILE;
        if (nb < NT) {
            const int line = (lane & 15) * 64;
            const char* s = (lane < 16) ? (const char*)(Temp + nb)
                                        : (const char*)(e1p + nb);
            __builtin_prefetch(s + line, 0, 0);
            if (lane < 16)
                __builtin_prefetch((const char*)(e2p + nb) + line, 0, 0);
        }

        __syncthreads();  // single wave: cheap; ensures LDS visibility (dscnt)

        // ---- phase 2: serial recurrence on lane 0, fed from LDS ----
        if (lane == 0) {
            float xl = x, yl = y;
            #pragma unroll 8
            for (int j = 0; j < nsteps; ++j) {
                const float4 c = coef[j];  // ds_load_b128, off the dep chain
                // critical path: 2 fma -> v_exp_f32 -> fma   (x2 with ILP)
                const float arg1 = fmaf(ab1, xl, fmaf(ag1, yl, c.x));
                const float arg2 = fmaf(ab2, yl, fmaf(ag2, xl, c.y));
                const float ex1 = __builtin_amdgcn_exp2f(arg1);
                const float ex2 = __builtin_amdgcn_exp2f(arg2);
                xl = fmaf(xl, ex1, c.z);
                yl = fmaf(yl, ex2, c.w);
                out[base + j + 1]     = xl;  // fire-and-forget stores
                out[n + base + j + 1] = yl;
            }
            x = xl;
            y = yl;
        }

        __syncthreads();
    }
}

extern "C" void kernel_launch(void* const* d_in, const int* in_sizes, int n_in,
                              void* d_out, int out_size, void* d_ws, size_t ws_size,
                              hipStream_t stream) {
    const float* Temp   = (const float*)d_in[0];  // (T_LEN,) float32
    const float* params = (const float*)d_in[1];  // (11,)    float32
    const float* eps    = (const float*)d_in[2];  // (2, T_LEN-1) float32, flat
    float* out = (float*)d_out;                   // (2, T_LEN) float32, flat
    const int n = in_sizes[0];

    // Serial recurrence: exactly one wave32 does the work.
    ricker_scan_kernel<<<dim3(1), dim3(32), 0, stream>>>(Temp, params, eps, out, n);
}